// GQNN_R_11914239279498
// MI455X (gfx1250) — compile-verified
//
#include <hip/hip_runtime.h>

#define N_NODES_C 100000
#define IN_DIM 64
#define HID 64
#define KPAD 68   // 65 (x|tau) padded to a multiple of 4 -> aligned rows, no WMMA K-tail

typedef __attribute__((ext_vector_type(2))) float v2f;
typedef __attribute__((ext_vector_type(8))) float v8f;

// Build padded augmented features h0 = [x | tau | 0 0 0]; zero both edge-sum
// buffers and deg. One thread per (node, col).
__global__ void k_prep(const float* __restrict__ x, const float* __restrict__ tau,
                       float* __restrict__ h0, float* __restrict__ sum1,
                       float* __restrict__ sum2, float* __restrict__ deg, int n)
{
    int idx = blockIdx.x * blockDim.x + threadIdx.x;
    int node = idx / KPAD;
    int c = idx - node * KPAD;
    if (node >= n) return;
    float v = 0.0f;
    if (c < IN_DIM)       v = x[node * IN_DIM + c];
    else if (c == IN_DIM) v = tau[node];
    h0[idx]   = v;
    sum1[idx] = 0.0f;
    if (c < HID)  sum2[node * HID + c] = 0.0f;
    if (c == 0)   deg[node] = 0.0f;
}

// Zero-pad the 65x64 layer-1 weights to 68x64 so the GEMM K-loop is uniform.
__global__ void k_padw(const float* __restrict__ Wl, const float* __restrict__ Wr,
                       float* __restrict__ Wlp, float* __restrict__ Wrp)
{
    int idx = blockIdx.x * blockDim.x + threadIdx.x;
    if (idx >= KPAD * HID) return;
    int k = idx / HID;
    float a = 0.0f, b = 0.0f;
    if (k < IN_DIM + 1) { a = Wl[idx]; b = Wr[idx]; }
    Wlp[idx] = a;
    Wrp[idx] = b;
}

// In-degree (float) via L2 atomics; one thread per edge.
__global__ void k_deg(const int* __restrict__ dst, float* __restrict__ deg, int n_edges)
{
    int e = blockIdx.x * blockDim.x + threadIdx.x;
    if (e >= n_edges) return;
    atomicAdd(&deg[dst[e]], 1.0f);
}

// deg -> 1/max(deg,1), consumed inside the GEMM A-fragment load (exact fold).
__global__ void k_rdeg(float* __restrict__ deg, int n)
{
    int i = blockIdx.x * blockDim.x + threadIdx.x;
    if (i >= n) return;
    deg[i] = 1.0f / fmaxf(deg[i], 1.0f);
}

// Edge aggregation: one wave32 per edge. Lanes stride the feature columns ->
// coalesced 256B gather from L2-resident features + per-float atomic scatter.
__global__ void k_aggr(const float* __restrict__ feat, int stride, int ncols,
                       const int* __restrict__ src, const int* __restrict__ dst,
                       float* __restrict__ accum, int n_edges)
{
    int lane = threadIdx.x & 31;
    int wid  = (blockIdx.x * blockDim.x + threadIdx.x) >> 5;
    if (wid >= n_edges) return;
    int s = src[wid];
    int d = dst[wid];
    const float* fp = feat + (size_t)s * stride;
    float*       ap = accum + (size_t)d * stride;
    for (int f = lane; f < ncols; f += 32)
        atomicAdd(ap + f, fp[f]);
}

// Fused SAGE layer: out = relu( (sum*rdeg) @ Bl + Aself @ Br + bias )
// One wave computes a 16x64 tile via V_WMMA_F32_16X16X4_F32.
// Weights staged in LDS in *fragment order*: pair (B[2kp][n], B[2kp+1][n])
// stored contiguously so each B fragment is a single aligned ds_load_b64.
// K is a template constant (68 or 64) -> fully unrolled straight-line WMMA chain.
template <int K>
__global__ void __launch_bounds__(256)
k_sage_gemm(const float* __restrict__ Asum, const float* __restrict__ rdeg,
            const float* __restrict__ Aself,
            const float* __restrict__ Bl, const float* __restrict__ Br,
            const float* __restrict__ bias, float* __restrict__ out,
            int n_rows)
{
    __shared__ float sBl[K * HID];
    __shared__ float sBr[K * HID];
    int tid = threadIdx.x;
    // swizzled staging: element (k, col) -> pair-major position
    for (int i = tid; i < K * HID; i += 256) {
        int k   = i >> 6;
        int col = i & 63;
        int pos = ((k >> 1) << 7) + (col << 1) + (k & 1);
        sBl[pos] = Bl[i];
        sBr[pos] = Br[i];
    }
    __syncthreads();

    int wave = tid >> 5;
    int lane = tid & 31;
    int half = lane >> 4;   // 0: K pair {0,1}, 1: K pair {2,3}
    int l15  = lane & 15;
    int row0 = (blockIdx.x * 8 + wave) * 16;
    if (row0 >= n_rows) return;   // no barriers after this point

    int arow = row0 + l15;        // every A-fragment element of this lane is row arow
    float r = rdeg[arow];
    const float* a1p = Asum  + (size_t)arow * K;
    const float* a2p = Aself + (size_t)arow * K;

    const v2f* blp = (const v2f*)sBl;   // pair-indexed: [kp * 64 + col]
    const v2f* brp = (const v2f*)sBr;

    v8f acc[4] = {};

    #pragma unroll
    for (int k0 = 0; k0 < K; k0 += 4) {
        int ka = k0 + 2 * half;       // this lane's K pair start
        int kp = (k0 >> 1) + half;    // pair index into swizzled LDS
        v2f a1 = *(const v2f*)(a1p + ka);
        a1 *= r;                      // fold mean = sum * (1/deg)
        v2f a2 = *(const v2f*)(a2p + ka);
        #pragma unroll
        for (int t = 0; t < 4; ++t) {
            int col = t * 16 + l15;
            v2f b1 = blp[kp * 64 + col];   // single ds_load_b64
            v2f b2 = brp[kp * 64 + col];
            acc[t] = __builtin_amdgcn_wmma_f32_16x16x4_f32(
                false, a1, false, b1, (short)0, acc[t], false, false);
            acc[t] = __builtin_amdgcn_wmma_f32_16x16x4_f32(
                false, a2, false, b2, (short)0, acc[t], false, false);
        }
    }

    // Epilogue: bias + ReLU + store (C/D layout: lanes 0-15 M=vgpr, 16-31 M=vgpr+8)
    #pragma unroll
    for (int t = 0; t < 4; ++t) {
        int col = t * 16 + l15;
        float bv = bias[col];
        #pragma unroll
        for (int i = 0; i < 8; ++i) {
            int m = i + 8 * half;
            float v = acc[t][i] + bv;
            v = fmaxf(v, 0.0f);
            out[(size_t)(row0 + m) * HID + col] = v;
        }
    }
}

// Final fc: one wave per node, 64-wide dot product + shuffle reduction.
__global__ void k_fc(const float* __restrict__ h, const float* __restrict__ Wfc,
                     const float* __restrict__ bfc, float* __restrict__ out, int n)
{
    int lane = threadIdx.x & 31;
    int node = (blockIdx.x * blockDim.x + threadIdx.x) >> 5;
    if (node >= n) return;
    const float* hp = h + (size_t)node * HID;
    float s = hp[lane] * Wfc[lane] + hp[lane + 32] * Wfc[lane + 32];
    #pragma unroll
    for (int off = 16; off > 0; off >>= 1)
        s += __shfl_down(s, off, 32);
    if (lane == 0) out[node] = s + bfc[0];
}

extern "C" void kernel_launch(void* const* d_in, const int* in_sizes, int n_in,
                              void* d_out, int out_size, void* d_ws, size_t ws_size,
                              hipStream_t stream)
{
    const float* x   = (const float*)d_in[0];
    const int*   ei  = (const int*)d_in[1];
    const float* tau = (const float*)d_in[2];
    const float* W1l = (const float*)d_in[3];
    const float* b1l = (const float*)d_in[4];
    const float* W1r = (const float*)d_in[5];
    const float* W2l = (const float*)d_in[6];
    const float* b2l = (const float*)d_in[7];
    const float* W2r = (const float*)d_in[8];
    const float* Wfc = (const float*)d_in[9];
    const float* bfc = (const float*)d_in[10];
    float* outp = (float*)d_out;

    const int n = N_NODES_C;
    const int n_edges = in_sizes[1] / 2;
    const int* src = ei;
    const int* dst = ei + n_edges;

    float* ws = (float*)d_ws;
    float* h0   = ws; ws += (size_t)n * KPAD;
    float* sum1 = ws; ws += (size_t)n * KPAD;
    float* h1   = ws; ws += (size_t)n * HID;
    float* sum2 = ws; ws += (size_t)n * HID;
    float* h2   = ws; ws += (size_t)n * HID;
    float* deg  = ws; ws += n;
    float* W1lp = ws; ws += KPAD * HID;
    float* W1rp = ws; ws += KPAD * HID;

    dim3 blk(256);

    k_prep<<<dim3((n * KPAD + 255) / 256), blk, 0, stream>>>(x, tau, h0, sum1, sum2, deg, n);
    k_padw<<<dim3((KPAD * HID + 255) / 256), blk, 0, stream>>>(W1l, W1r, W1lp, W1rp);
    k_deg<<<dim3((n_edges + 255) / 256), blk, 0, stream>>>(dst, deg, n_edges);
    k_aggr<<<dim3((n_edges + 7) / 8), blk, 0, stream>>>(h0, KPAD, IN_DIM + 1, src, dst, sum1, n_edges);
    k_rdeg<<<dim3((n + 255) / 256), blk, 0, stream>>>(deg, n);
    k_sage_gemm<KPAD><<<dim3((n + 127) / 128), blk, 0, stream>>>(sum1, deg, h0, W1lp, W1rp, b1l, h1, n);
    k_aggr<<<dim3((n_edges + 7) / 8), blk, 0, stream>>>(h1, HID, HID, src, dst, sum2, n_edges);
    k_sage_gemm<HID><<<dim3((n + 127) / 128), blk, 0, stream>>>(sum2, deg, h1, W2l, W2r, b2l, h2, n);
    k_fc<<<dim3((n * 32 + 255) / 256), blk, 0, stream>>>(h2, Wfc, bfc, outp, n);
}